// GCN_335007449371
// MI455X (gfx1250) — compile-verified
//
#include <hip/hip_runtime.h>
#include <math.h>

#define NN 100000
#define NE 1600000

typedef __attribute__((ext_vector_type(2))) float v2f;
typedef __attribute__((ext_vector_type(8))) float v8f;
typedef __attribute__((ext_vector_type(4))) unsigned v4u;
typedef __attribute__((ext_vector_type(8))) int v8i;
typedef __attribute__((ext_vector_type(4))) int v4i;

#if defined(__has_builtin)
#if __has_builtin(__builtin_amdgcn_tensor_load_to_lds)
#define HAVE_TDM 1
#endif
#endif
#ifndef HAVE_TDM
#define HAVE_TDM 0
#endif

// ---------------- degree / normalization ----------------

__global__ void zero_kernel(float* __restrict__ p, int n) {
  int i = blockIdx.x * blockDim.x + threadIdx.x;
  if (i < n) p[i] = 0.0f;
}

__global__ void deg_kernel(const int* __restrict__ dst, float* __restrict__ deg) {
  int e = blockIdx.x * blockDim.x + threadIdx.x;
  if (e < NE) atomicAdd(&deg[dst[e]], 1.0f);
}

__global__ void dinv_kernel(float* __restrict__ deg) {
  int i = blockIdx.x * blockDim.x + threadIdx.x;
  if (i < NN) deg[i] = rsqrtf(1.0f + deg[i]);
}

// ---------------- WMMA f32 GEMM:  C[N,F] = A[N,128] @ W[128,F] ----------------
// Block = 256 threads = 8 wave32. One 16-row tile per block, one 16-col tile per wave.
// W[128,F] staged into LDS by the Tensor Data Mover (one TENSOR_LOAD_TO_LDS per
// block, issued by wave 0; OOB reads beyond tensor_dim0=F return zero, which
// zero-pads the F=40 case to FPAD=48 for free). Math via V_WMMA_F32_16X16X4_F32
// (exact fp32, matches the fp32 reference; GEMM is not the bottleneck anyway —
// edge aggregation is, and its 51MB working set lives in the 192MB L2).

template<int F, int FPAD>
__global__ __launch_bounds__(256) void gemm_wmma(const float* __restrict__ A,
                                                 const float* __restrict__ W,
                                                 float* __restrict__ C, int nrows) {
  __shared__ float wlds[128 * FPAD];

#if HAVE_TDM
  if (threadIdx.x < 32) {   // one TDM issue per workgroup (wave 0)
    unsigned long long ga = (unsigned long long)(uintptr_t)W;
    v4u g0;
    g0.x = 1u;                                         // count=1 (valid user D#)
    g0.y = 0u;                                         // lds_addr = 0 (wlds base)
    g0.z = (unsigned)(ga & 0xFFFFFFFFu);               // global_addr[31:0]
    g0.w = (unsigned)((ga >> 32) & 0x1FFFFFFu)         // global_addr[56:32]
         | (2u << 30);                                 // type = 2 ("image")
    v8i g1;
    g1[0] = 0x20000;                                   // data_size=2 -> 4 bytes
    g1[1] = (int)(((unsigned)F & 0xFFFFu) << 16);      // tensor_dim0[15:0]
    g1[2] = (int)(((unsigned)F >> 16) | (128u << 16)); // tdim0[31:16] | tensor_dim1=128
    g1[3] = (int)((unsigned)FPAD << 16);               // tdim1[31:16]=0 | tile_dim0=FPAD
    g1[4] = 128;                                       // tile_dim1=128, tile_dim2=0
    g1[5] = F;                                         // tensor_dim0_stride[31:0]
    g1[6] = 0;                                         // stride0 hi | stride1 lo
    g1[7] = 0;                                         // stride1 hi
    v4i z4 = {};
#if __clang_major__ >= 23
    v8i z8 = {};
    __builtin_amdgcn_tensor_load_to_lds(g0, g1, z4, z4, z8, 0);
#else
    __builtin_amdgcn_tensor_load_to_lds(g0, g1, z4, z4, 0);
#endif
    __builtin_amdgcn_s_wait_tensorcnt(0);
  }
  __syncthreads();
#else
  for (int idx = threadIdx.x; idx < 128 * FPAD; idx += 256) {
    int k = idx / FPAD, c = idx - k * FPAD;
    wlds[idx] = (c < F) ? W[k * F + c] : 0.0f;
  }
  __syncthreads();
#endif

  const int wave = threadIdx.x >> 5;
  const int lane = threadIdx.x & 31;
  const int col0 = wave * 16;
  if (col0 >= FPAD) return;          // uniform per wave (only for F=40 case)

  const int m    = lane & 15;
  const int koff = (lane >> 4) << 1; // lanes 0-15 -> K+{0,1}; lanes 16-31 -> K+{2,3}
  const int row0 = blockIdx.x * 16;
  int ar = row0 + m; if (ar >= nrows) ar = 0;   // N%16==0, but stay safe
  const float* arow = A + (size_t)ar * 128;

  v8f acc = {};
#pragma unroll
  for (int k = 0; k < 128; k += 4) {
    v2f a, b;
    a.x = arow[k + koff];
    a.y = arow[k + koff + 1];
    b.x = wlds[(k + koff) * FPAD + col0 + m];
    b.y = wlds[(k + koff + 1) * FPAD + col0 + m];
    acc = __builtin_amdgcn_wmma_f32_16x16x4_f32(false, a, false, b,
                                                (short)0, acc, false, false);
  }

  // C/D layout: lanes 0-15 hold M=v, lanes 16-31 hold M=8+v; N = lane&15
  const int rbase = row0 + ((lane >> 4) << 3);
  const int ccol  = col0 + m;
#pragma unroll
  for (int v = 0; v < 8; ++v) {
    int r = rbase + v;
    if (r < nrows && ccol < F) C[(size_t)r * F + ccol] = acc[v];
  }
}

// ---------------- GCN aggregation ----------------

template<int F>
__global__ void agg_init(const float* __restrict__ hw, const float* __restrict__ dinv,
                         const float* __restrict__ bias, float* __restrict__ out) {
  int i = blockIdx.x * blockDim.x + threadIdx.x;
  if (i >= NN * F) return;
  int node = i / F, f = i - node * F;
  float sn = dinv[node]; sn *= sn;               // self_norm = d^-1
  out[i] = hw[i] * sn + bias[f];
}

template<int F>
__global__ void agg_edges(const float* __restrict__ hw, const int* __restrict__ srcv,
                          const int* __restrict__ dstv, const float* __restrict__ dinv,
                          float* __restrict__ out) {
  constexpr int CH = F / 4;                      // float4 chunks per row
  int t = blockIdx.x * blockDim.x + threadIdx.x;
  int e = t / CH;
  if (e >= NE) return;
  int c = t - e * CH;
  int s = srcv[e], d = dstv[e];
  float ne = dinv[s] * dinv[d];
  const float4 hv = *(const float4*)(hw + (size_t)s * F + c * 4);
  float* o = out + (size_t)d * F + c * 4;        // 51MB working set lives in L2
  atomicAdd(o + 0, hv.x * ne);
  atomicAdd(o + 1, hv.y * ne);
  atomicAdd(o + 2, hv.z * ne);
  atomicAdd(o + 3, hv.w * ne);
}

// ---------------- batchnorm (training stats) + ReLU ----------------

__global__ __launch_bounds__(128) void bn_stats(const float* __restrict__ h,
                                                float* __restrict__ stats) {
  int f = threadIdx.x;                           // column, coalesced across lanes
  float s = 0.f, sq = 0.f;
  for (int r = blockIdx.x; r < NN; r += gridDim.x) {
    float v = h[(size_t)r * 128 + f];
    s += v; sq += v * v;
  }
  atomicAdd(&stats[f], s);
  atomicAdd(&stats[128 + f], sq);
}

__global__ void bn_apply(float* __restrict__ h, const float* __restrict__ stats,
                         const float* __restrict__ g, const float* __restrict__ be) {
  int i = blockIdx.x * blockDim.x + threadIdx.x;
  if (i >= NN * 128) return;
  int f = i & 127;
  const float inv_n = 1.0f / (float)NN;
  float mu  = stats[f] * inv_n;
  float var = stats[128 + f] * inv_n - mu * mu;  // biased var = E[x^2]-mu^2
  float v = (h[i] - mu) * rsqrtf(var + 1e-5f) * g[f] + be[f];
  h[i] = fmaxf(v, 0.0f);
}

// ---------------- log_softmax over 40 classes, one wave32 per row ----------------

__global__ __launch_bounds__(256) void log_softmax40(const float* __restrict__ h,
                                                     float* __restrict__ out) {
  int row  = blockIdx.x * 8 + (threadIdx.x >> 5);
  int lane = threadIdx.x & 31;
  if (row >= NN) return;
  const float* hr = h + (size_t)row * 40;
  float v0 = hr[lane];
  float v1 = (lane < 8) ? hr[lane + 32] : -3.4e38f;
  float m = fmaxf(v0, v1);
  for (int o = 16; o; o >>= 1) m = fmaxf(m, __shfl_xor(m, o, 32));
  float e = expf(v0 - m) + ((lane < 8) ? expf(v1 - m) : 0.0f);
  for (int o = 16; o; o >>= 1) e += __shfl_xor(e, o, 32);
  float lse = m + logf(e);
  float* orow = out + (size_t)row * 40;
  orow[lane] = v0 - lse;
  if (lane < 8) orow[lane + 32] = v1 - lse;
}

// ---------------- host orchestration ----------------

extern "C" void kernel_launch(void* const* d_in, const int* in_sizes, int n_in,
                              void* d_out, int out_size, void* d_ws, size_t ws_size,
                              hipStream_t stream) {
  const float* x   = (const float*)d_in[0];
  const int*   src = (const int*)  d_in[1];
  const int*   dst = (const int*)  d_in[2];
  const float* W0  = (const float*)d_in[3];
  const float* b0  = (const float*)d_in[4];
  const float* W1  = (const float*)d_in[5];
  const float* b1  = (const float*)d_in[6];
  const float* W2  = (const float*)d_in[7];
  const float* b2  = (const float*)d_in[8];
  const float* g0  = (const float*)d_in[9];
  const float* be0 = (const float*)d_in[10];
  const float* g1  = (const float*)d_in[11];
  const float* be1 = (const float*)d_in[12];
  float* out = (float*)d_out;

  // workspace layout (floats): dinv[100096] | stats[256] | buf0[N*128] | buf1[N*128]
  float* ws    = (float*)d_ws;
  float* dinv  = ws;
  float* stats = ws + 100096;
  float* buf0  = ws + 100352;
  float* buf1  = buf0 + (size_t)NN * 128;

  const int B = 256;
  // degree -> d^-1/2
  zero_kernel<<<(NN + B - 1) / B, B, 0, stream>>>(dinv, NN);
  deg_kernel<<<(NE + B - 1) / B, B, 0, stream>>>(dst, dinv);
  dinv_kernel<<<(NN + B - 1) / B, B, 0, stream>>>(dinv);

  const int gemm_blocks = NN / 16;          // 6250, exact
  // ---- layer 0 ----
  gemm_wmma<128, 128><<<gemm_blocks, 256, 0, stream>>>(x, W0, buf0, NN);
  agg_init<128><<<(NN * 128) / B, B, 0, stream>>>(buf0, dinv, b0, buf1);
  agg_edges<128><<<(NE * 32) / B, B, 0, stream>>>(buf0, src, dst, dinv, buf1);
  zero_kernel<<<1, 256, 0, stream>>>(stats, 256);
  bn_stats<<<1024, 128, 0, stream>>>(buf1, stats);
  bn_apply<<<(NN * 128) / B, B, 0, stream>>>(buf1, stats, g0, be0);

  // ---- layer 1 ----
  gemm_wmma<128, 128><<<gemm_blocks, 256, 0, stream>>>(buf1, W1, buf0, NN);
  agg_init<128><<<(NN * 128) / B, B, 0, stream>>>(buf0, dinv, b1, buf1);
  agg_edges<128><<<(NE * 32) / B, B, 0, stream>>>(buf0, src, dst, dinv, buf1);
  zero_kernel<<<1, 256, 0, stream>>>(stats, 256);
  bn_stats<<<1024, 128, 0, stream>>>(buf1, stats);
  bn_apply<<<(NN * 128) / B, B, 0, stream>>>(buf1, stats, g1, be1);

  // ---- layer 2 (F=40, padded to 48 col-tiles; TDM zero-pads W2) ----
  gemm_wmma<40, 48><<<gemm_blocks, 256, 0, stream>>>(buf1, W2, buf0, NN);
  agg_init<40><<<(NN * 40 + B - 1) / B, B, 0, stream>>>(buf0, dinv, b2, buf1);
  agg_edges<40><<<(NE * 10 + B - 1) / B, B, 0, stream>>>(buf0, src, dst, dinv, buf1);

  log_softmax40<<<(NN + 7) / 8, 256, 0, stream>>>(buf1, out);
}